// ShiftConv2d_86973087744740
// MI455X (gfx1250) — compile-verified
//
#include <hip/hip_runtime.h>

typedef __attribute__((ext_vector_type(2))) float v2f;
typedef __attribute__((ext_vector_type(8))) float v8f;
typedef int v4i __attribute__((vector_size(16)));   // matches builtin's int4

#define CIN   256
#define COUT  256
#define HH    128
#define WW    128
#define GRP   51      // CIN / 5
#define YS    136     // LDS row stride (floats): 2*YS mod 64 == 16 -> lane halves on disjoint banks
#define GRIDX 512     // persistent blocks; 2048 rows -> 4 rows per block

#if __has_builtin(__builtin_amdgcn_global_load_async_to_lds_b128)
#define HAVE_ASYNC 1
#else
#define HAVE_ASYNC 0
#endif

#if __has_builtin(__builtin_amdgcn_s_wait_asynccnt)
#define WAITASYNC(n) __builtin_amdgcn_s_wait_asynccnt(n)
#else
#define WAITASYNC(n) asm volatile("s_wait_asynccnt %0" ::"n"(n) : "memory")
#endif

typedef __attribute__((address_space(1))) v4i g_v4i;   // global int4
typedef __attribute__((address_space(3))) v4i l_v4i;   // LDS int4

__device__ __forceinline__ void chan_shift(int c, int& dh, int& dw) {
    dh = 0; dw = 0;
    if      (c <     GRP) dw =  2;   // y[h,w] = x[h, w+2]
    else if (c < 2 * GRP) dw = -2;   // x[h, w-2]
    else if (c < 3 * GRP) dh =  2;   // x[h+2, w]
    else if (c < 4 * GRP) dh = -2;   // x[h-2, w]
}

// Issue staging for one (b,h) row into Ybuf. Async path: 16 x b128 DMA per wave,
// raw shifted addresses (always inside the x allocation); edges fixed later.
__device__ __forceinline__ void stage_row(const float* __restrict__ x, float* Ybuf,
                                          int b, int h, int wave, int lane) {
    for (int it = 0; it < 16; ++it) {
        const int c = wave + (it << 4);
        int dh, dw; chan_shift(c, dh, dw);
        const int hs = h + dh;
#if HAVE_ASYNC
        const int w0 = lane << 2;   // 4 floats per lane, 128 per wave-channel
        const float* gsrc = x + (((b * CIN + c) * HH + hs) * WW + w0 + dw);
        float*       ldst = Ybuf + c * YS + w0;
        __builtin_amdgcn_global_load_async_to_lds_b128((g_v4i*)gsrc, (l_v4i*)ldst, 0, 0);
#else
        const bool hok = (unsigned)hs < (unsigned)HH;
        const float* src = x + (((b * CIN + c) * HH + hs) * WW);
#pragma unroll
        for (int j = 0; j < 4; ++j) {
            const int wcol = lane + (j << 5);
            const int wsrc = wcol + dw;
            float v = (hok && (unsigned)wsrc < (unsigned)WW) ? src[wsrc] : 0.0f;
            Ybuf[c * YS + wcol] = v;
        }
#endif
    }
}

#if HAVE_ASYNC
// Zero the shift halos the raw DMA loads filled with neighbor data.
__device__ __forceinline__ void fix_edges(float* Ybuf, int h, int wave, int lane) {
    for (int it = 0; it < 16; ++it) {
        const int c = wave + (it << 4);
        int dh, dw; chan_shift(c, dh, dw);
        const int hs = h + dh;
        if ((unsigned)hs >= (unsigned)HH) {
            // whole row out of image -> zero 128 floats (b128 store per lane)
            float4 z = {0.f, 0.f, 0.f, 0.f};
            *(float4*)&Ybuf[c * YS + (lane << 2)] = z;
        } else if (dw == 2) {
            if (lane < 2) Ybuf[c * YS + 126 + lane] = 0.f;
        } else if (dw == -2) {
            if (lane < 2) Ybuf[c * YS + lane] = 0.f;
        }
    }
}
#endif

__global__ __launch_bounds__(512, 1)
void shift_conv1x1_wmma(const float* __restrict__ x,
                        const float* __restrict__ wmat,
                        const float* __restrict__ bias,
                        float* __restrict__ out,
                        int totalRows) {
    __shared__ float Ysh[2][CIN * YS];   // 2 x 139,264 B = 272 KB (<= 320 KB/WGP)

    const int tid   = threadIdx.x;
    const int lane  = tid & 31;
    const int wave  = tid >> 5;          // 0..15
    const int lhalf = lane >> 4;
    const int lrow  = lane & 15;
    const int mWave = wave << 4;

    // Per-wave invariants (persist across rows)
    const float* wrow = wmat + (mWave + lrow) * CIN + 2 * lhalf;
    float bv[8];
#pragma unroll
    for (int v = 0; v < 8; ++v) bv[v] = bias[mWave + 8 * lhalf + v];

    const int r0     = blockIdx.x;
    const int stride = gridDim.x;
    if (r0 >= totalRows) return;

    // Prologue: stage first row into buffer 0
    stage_row(x, Ysh[0], r0 >> 7, r0 & 127, wave, lane);

    int buf = 0;
    for (int r = r0; r < totalRows; r += stride) {
        const int b = r >> 7;
        const int h = r & 127;

        // Stage next row into the other buffer (overlaps with this row's WMMA loop)
        const int rn = r + stride;
        if (rn < totalRows) {
            stage_row(x, Ysh[buf ^ 1], rn >> 7, rn & 127, wave, lane);
#if HAVE_ASYNC
            WAITASYNC(16);   // older 16 DMAs (current buffer) complete; newer may fly
#endif
        } else {
#if HAVE_ASYNC
            WAITASYNC(0);
#endif
        }
#if HAVE_ASYNC
        fix_edges(Ysh[buf], h, wave, lane);
#endif
        __syncthreads();

        // ---------- GEMM: wave's 16 output channels x 128 pixels ----------
        const float* Ybuf = Ysh[buf];
        v8f acc[8];
#pragma unroll
        for (int v = 0; v < 8; ++v)
#pragma unroll
            for (int t = 0; t < 8; ++t) acc[t][v] = bv[v];

#pragma unroll 2
        for (int k = 0; k < CIN; k += 4) {
            v2f a;                       // A 16x4 frag: one b64 per lane
            a.x = wrow[k];
            a.y = wrow[k + 1];
            const float* yrow = &Ybuf[(k + 2 * lhalf) * YS + lrow];
#pragma unroll
            for (int t = 0; t < 8; ++t) {
                v2f bb;                  // B 4x16 frag
                bb.x = yrow[t * 16];
                bb.y = yrow[t * 16 + YS];
                acc[t] = __builtin_amdgcn_wmma_f32_16x16x4_f32(
                    false, a, false, bb, (short)0, acc[t], false, false);
            }
        }

        // ---------- Store out[b][o][h][w] ----------
        float* orow = out + (((b * COUT + mWave + 8 * lhalf) * HH + h) * WW);
#pragma unroll
        for (int v = 0; v < 8; ++v) {
            float* op = orow + v * (HH * WW);
#pragma unroll
            for (int t = 0; t < 8; ++t) op[t * 16 + lrow] = acc[t][v];
        }

        __syncthreads();   // all waves done reading buf before next iter DMAs into it
        buf ^= 1;
    }
}

extern "C" void kernel_launch(void* const* d_in, const int* in_sizes, int n_in,
                              void* d_out, int out_size, void* d_ws, size_t ws_size,
                              hipStream_t stream) {
    const float* x    = (const float*)d_in[0];
    const float* w1x1 = (const float*)d_in[1];
    const float* b1x1 = (const float*)d_in[2];
    float* out = (float*)d_out;

    const int B = in_sizes[0] / (CIN * HH * WW);   // 16
    const int totalRows = B * HH;                  // 2048
    const int grid = (totalRows < GRIDX) ? totalRows : GRIDX;
    hipLaunchKernelGGL(shift_conv1x1_wmma, dim3(grid), dim3(512), 0, stream,
                       x, w1x1, b1x1, out, totalRows);
}